// PredictionModel_28192165331192
// MI455X (gfx1250) — compile-verified
//
#include <hip/hip_runtime.h>
#include <math.h>

// ---------------------------------------------------------------------------
// Problem constants (from reference)
// ---------------------------------------------------------------------------
#define NN    32      // N
#define KK    16      // K
#define BINS  100
#define DIMD  2
#define NB    102     // 101 (BINS+1) padded to even for round-robin Jacobi
#define PB    112     // 101 padded to multiple of 16 for WMMA
#define PT    128     // max T (sample_size-1) padded to multiple of 16
#define JITTER    1e-5f
#define EPSD      1e-12f

// ---------------------------------------------------------------------------
// Workspace layout (float offsets)
// ---------------------------------------------------------------------------
#define OFF_AB  0                       // Bk'  padded PB x PB
#define OFF_UB  (OFF_AB + PB*PB)        // U_B  padded PB x PB (eigvecs as cols)
#define OFF_DB  (OFF_UB + PB*PB)        // d_B  PB
#define OFF_AC  (OFF_DB + PB)           // CC^T 32 x 32
#define OFF_UC  (OFF_AC + NN*NN)        // U_C  32 x 32
#define OFF_DC  (OFF_UC + NN*NN)        // d_C  32
#define OFF_X   (OFF_DC + NN)           // X    2 x PB x 32
#define OFF_M1  (OFF_X  + 2*PB*NN)      // M1 = U_B^T X        2 x PB x 32
#define OFF_Z   (OFF_M1 + 2*PB*NN)      // Z  = M1 U_C         2 x PB x 32
#define OFF_W2  (OFF_Z  + 2*PB*NN)      // W2 = Z*dC/(s2+dBdC) 2 x PB x 32
#define OFF_Y2  (OFF_W2 + 2*PB*NN)      // Y2 = U_B W2         2 x PB x 32
#define OFF_M2  (OFF_Y2 + 2*PB*NN)      // M2 = Y2 U_C^T       2 x PB x 32
#define OFF_BC  (OFF_M2 + 2*PB*NN)      // B_cross padded PB x PT
#define OFF_P   (OFF_BC + PB*PT)        // pred_v  2 x PT x 32
#define OFF_XL  (OFF_P  + 2*PT*NN)      // x_last  32 x 2
#define OFF_XT  (OFF_XL + NN*DIMD)      // xt      PT x 32 x 2
#define WS_FLOATS (OFF_XT + PT*NN*DIMD)

typedef __attribute__((ext_vector_type(2))) float v2f;
typedef __attribute__((ext_vector_type(8))) float v8f;

// ---------------------------------------------------------------------------
// 0) zero workspace (harness poisons it with 0xAA)
// ---------------------------------------------------------------------------
__global__ void zero_ws_kernel(float* ws) {
  int i = blockIdx.x * blockDim.x + threadIdx.x;
  if (i < WS_FLOATS) ws[i] = 0.0f;
}

// ---------------------------------------------------------------------------
// 1) Build Bk', CC^T, identities, X, x_last, B_cross
// ---------------------------------------------------------------------------
__global__ __launch_bounds__(256) void build_kernel(
    const float* __restrict__ x0, const float* __restrict__ v,
    const float* __restrict__ x0c, const float* __restrict__ ls,
    const float* __restrict__ CQ, const int* __restrict__ ssz,
    float* __restrict__ ws) {
  const int tid   = threadIdx.x;
  const float l   = ls[0];
  const float inv2l2 = 0.5f / (l * l);
  const int S = ssz[0];
  const int T = (S - 1 < PT) ? (S - 1) : PT;
  const float step = 1.0f / (float)(S - 1);

  // Bk' (102x102 live region of PB x PB)
  for (int idx = tid; idx < PB * PB; idx += 256) {
    int p = idx / PB, q = idx % PB;
    float val = 0.0f;
    if (p == 0 && q == 0) {
      val = x0c[0] * x0c[0] + JITTER;
    } else if (p >= 1 && p <= BINS && q >= 1 && q <= BINS) {
      float cp = ((float)(p - 1) + 0.5f) / (float)BINS;
      float cq = ((float)(q - 1) + 0.5f) / (float)BINS;
      float d  = cp - cq;
      val = __expf(-d * d * inv2l2);
      if (p == q) val += JITTER;
    }
    ws[OFF_AB + idx] = val;
  }
  // U_B, U_C identities
  for (int i = tid; i < NB; i += 256) ws[OFF_UB + i * PB + i] = 1.0f;
  for (int i = tid; i < NN; i += 256) ws[OFF_UC + i * NN + i] = 1.0f;
  // CC^T
  for (int idx = tid; idx < NN * NN; idx += 256) {
    int i = idx / NN, j = idx % NN;
    float s = 0.0f;
    for (int k = 0; k < KK; ++k) s += CQ[i * KK + k] * CQ[j * KK + k];
    ws[OFF_AC + idx] = s;
  }
  // X[d][p][n]: p==0 -> x0, p in 1..BINS -> v[p-1]
  for (int idx = tid; idx < DIMD * (BINS + 1) * NN; idx += 256) {
    int d = idx / ((BINS + 1) * NN);
    int r = idx % ((BINS + 1) * NN);
    int p = r / NN, n = r % NN;
    float val = (p == 0) ? x0[n * DIMD + d]
                         : v[((p - 1) * NN + n) * DIMD + d];
    ws[OFF_X + d * PB * NN + p * NN + n] = val;
  }
  // x_last = x0 + (1/BINS) * sum_b v
  for (int nd = tid; nd < NN * DIMD; nd += 256) {
    float s = 0.0f;
    for (int b = 0; b < BINS; ++b) s += v[b * NN * DIMD + nd];
    ws[OFF_XL + nd] = x0[nd] + s * (1.0f / (float)BINS);
  }
  // B_cross[p][t] = (p==0) ? 0 : rbf(ts[t], centers[p-1])
  for (int idx = tid; idx < PB * PT; idx += 256) {
    int p = idx / PT, t = idx % PT;
    float val = 0.0f;
    if (p >= 1 && p <= BINS && t < T) {
      float ts = 1.0f + (float)t * step;
      float c  = ((float)(p - 1) + 0.5f) / (float)BINS;
      float d  = ts - c;
      val = __expf(-d * d * inv2l2);
    }
    ws[OFF_BC + idx] = val;
  }
}

// ---------------------------------------------------------------------------
// 2) One-workgroup parallel cyclic Jacobi eigensolver.
//    block 0: 102x102 (Bk'), block 1: 32x32 (CC^T). A in LDS, V in global.
// ---------------------------------------------------------------------------
#define JACOBI_SWEEPS 8
__global__ __launch_bounds__(256) void jacobi_kernel(float* __restrict__ ws) {
  __shared__ float A[NB * NB];
  __shared__ float cs_c[NB / 2], cs_s[NB / 2];
  __shared__ int   cs_p[NB / 2], cs_q[NB / 2];

  const int which = blockIdx.x;
  const int n   = (which == 0) ? NB : NN;
  const int gld = (which == 0) ? PB : NN;
  float* gA = ws + ((which == 0) ? OFF_AB : OFF_AC);
  float* gV = ws + ((which == 0) ? OFF_UB : OFF_UC);
  float* gD = ws + ((which == 0) ? OFF_DB : OFF_DC);
  const int tid = threadIdx.x;
  const int npairs = n / 2;

  for (int idx = tid; idx < n * n; idx += 256)
    A[idx] = gA[(idx / n) * gld + (idx % n)];
  __syncthreads();

  for (int sweep = 0; sweep < JACOBI_SWEEPS; ++sweep) {
    for (int r = 0; r < n - 1; ++r) {
      // phase 1: rotation angles for the round-robin pairing
      if (tid < npairs) {
        int p, q;
        if (tid == 0) { p = n - 1; q = r % (n - 1); }
        else {
          p = (r + tid) % (n - 1);
          q = (r - tid + 2 * (n - 1)) % (n - 1);
        }
        if (p > q) { int t = p; p = q; q = t; }
        float apq = A[p * n + q];
        float c = 1.0f, s = 0.0f;
        if (fabsf(apq) > 1e-20f) {
          float tau = (A[q * n + q] - A[p * n + p]) / (2.0f * apq);
          float t = ((tau >= 0.0f) ? 1.0f : -1.0f) /
                    (fabsf(tau) + sqrtf(1.0f + tau * tau));
          c = 1.0f / sqrtf(1.0f + t * t);
          s = t * c;
        }
        cs_c[tid] = c; cs_s[tid] = s; cs_p[tid] = p; cs_q[tid] = q;
      }
      __syncthreads();
      // phase 2: column rotations on A (LDS) and V (global, disjoint columns)
      for (int idx = tid; idx < npairs * n; idx += 256) {
        int k = idx / n, i = idx % n;
        int p = cs_p[k], q = cs_q[k];
        float c = cs_c[k], s = cs_s[k];
        float aip = A[i * n + p], aiq = A[i * n + q];
        A[i * n + p] = c * aip - s * aiq;
        A[i * n + q] = s * aip + c * aiq;
        float vip = gV[i * gld + p], viq = gV[i * gld + q];
        gV[i * gld + p] = c * vip - s * viq;
        gV[i * gld + q] = s * vip + c * viq;
      }
      __threadfence_block();
      __syncthreads();
      // phase 3: row rotations on A (disjoint rows)
      for (int idx = tid; idx < npairs * n; idx += 256) {
        int k = idx / n, j = idx % n;
        int p = cs_p[k], q = cs_q[k];
        float c = cs_c[k], s = cs_s[k];
        float apj = A[p * n + j], aqj = A[q * n + j];
        A[p * n + j] = c * apj - s * aqj;
        A[q * n + j] = s * apj + c * aqj;
      }
      __syncthreads();
    }
  }
  for (int i = tid; i < n; i += 256) gD[i] = A[i * n + i];
}

// ---------------------------------------------------------------------------
// 3) Generic f32 WMMA GEMM, transpose flags as template params (branch-free
//    inner loop). One wave per 16x16 output tile; blockIdx.z = batch (channel)
//    with per-z operand strides. K-loop unrolled 4x -> 4 back-to-back
//    V_WMMA_F32_16X16X4_F32 per iteration. EXEC all-ones (32 threads).
// ---------------------------------------------------------------------------
template <bool TA, bool TB>
__global__ __launch_bounds__(32) void wmma_gemm_f32(
    const float* __restrict__ A, int lda, int strideAz,
    const float* __restrict__ B, int ldb, int strideBz,
    float* __restrict__ D, int ldd, int strideDz, int Ktot) {
  const int z = blockIdx.z;
  A += z * strideAz;
  B += z * strideBz;
  D += z * strideDz;

  const int lane = threadIdx.x;
  const int half = lane >> 4;          // 0: lanes 0-15, 1: lanes 16-31
  const int l    = lane & 15;
  const int m    = blockIdx.y * 16 + l;
  const int nc   = blockIdx.x * 16 + l;
  const int ka   = half * 2;           // A frag: VGPR0 = K={0,2}, VGPR1 = K={1,3}

  // gfx1250 prefetch of the operand rows this lane will walk
  __builtin_prefetch(TA ? (A + ka * lda + m) : (A + m * lda + ka), 0, 3);
  __builtin_prefetch(TB ? (B + nc * ldb + ka) : (B + ka * ldb + nc), 0, 3);

  v8f acc = {};
#pragma unroll 4
  for (int k0 = 0; k0 < Ktot; k0 += 4) {
    v2f a, b;
    if (TA) {                          // A^T: K-major -> strided gather
      a.x = A[(k0 + ka + 0) * lda + m];
      a.y = A[(k0 + ka + 1) * lda + m];
    } else {                           // row-major: contiguous pair -> b64 load
      a = *(const v2f*)(A + m * lda + (k0 + ka));
    }
    if (TB) {                          // B^T: contiguous pair -> b64 load
      b = *(const v2f*)(B + nc * ldb + (k0 + ka));
    } else {
      b.x = B[(k0 + ka + 0) * ldb + nc];
      b.y = B[(k0 + ka + 1) * ldb + nc];
    }
    acc = __builtin_amdgcn_wmma_f32_16x16x4_f32(
        /*neg_a=*/false, a, /*neg_b=*/false, b,
        /*c_mod=*/(short)0, acc, /*reuse_a=*/false, /*reuse_b=*/false);
  }
  // C/D layout: VGPR r -> M = 8*half + r, N = lane&15
  const int m0 = blockIdx.y * 16 + half * 8;
#pragma unroll
  for (int r = 0; r < 8; ++r) D[(m0 + r) * ldd + nc] = acc[r];
}

// ---------------------------------------------------------------------------
// 4) Elementwise spectral solve core: W2 = Z * dC / (sigma^2 + dB*dC)
// ---------------------------------------------------------------------------
__global__ void w2_kernel(const float* __restrict__ sigma, float* __restrict__ ws) {
  int idx = blockIdx.x * blockDim.x + threadIdx.x;
  if (idx >= 2 * PB * NN) return;
  int rem = idx % (PB * NN);
  int p = rem / NN, k = rem % NN;
  float sg = fmaxf(sigma[0], 5.0f / (float)BINS);
  float s2 = sg * sg;
  float dB = ws[OFF_DB + p];
  float dC = ws[OFF_DC + k];
  ws[OFF_W2 + idx] = ws[OFF_Z + idx] * dC / (s2 + dB * dC);
}

// ---------------------------------------------------------------------------
// 5) xt[t][n][d] = x_last[n][d] + (ts[t]-1) * pred_v[t][n][d]
// ---------------------------------------------------------------------------
__global__ void xt_kernel(const int* __restrict__ ssz, float* __restrict__ ws) {
  int idx = blockIdx.x * blockDim.x + threadIdx.x;
  const int S = ssz[0];
  const int T = (S - 1 < PT) ? (S - 1) : PT;
  if (idx >= T * NN * DIMD) return;
  int t = idx / (NN * DIMD);
  int nd = idx % (NN * DIMD);
  int n = nd / DIMD, d = nd % DIMD;
  float step = 1.0f / (float)(S - 1);
  float tsrel = (float)t * step;             // ts[t] - LAST_TIME
  float pv = ws[OFF_P + d * PT * NN + t * NN + n];
  ws[OFF_XT + idx] = ws[OFF_XL + nd] + tsrel * pv;
}

// ---------------------------------------------------------------------------
// 6) Intensity integral: out[i][j] = dt * sum_t exp(b_i+b_j - dist_t(i,j))
// ---------------------------------------------------------------------------
__global__ __launch_bounds__(1024) void intensity_kernel(
    const float* __restrict__ beta, const int* __restrict__ ssz,
    const float* __restrict__ ws, float* __restrict__ out) {
  __shared__ float sxt[PT * NN * DIMD];      // 32 KB
  const int S = ssz[0];
  const int T = (S - 1 < PT) ? (S - 1) : PT;
  const int tid = threadIdx.x;
  for (int idx = tid; idx < T * NN * DIMD; idx += 1024)
    sxt[idx] = ws[OFF_XT + idx];
  __syncthreads();
  const int i = tid / NN, j = tid % NN;
  const float bb = beta[i] + beta[j];
  float acc = 0.0f;
  for (int t = 0; t < T; ++t) {
    const float* row = &sxt[t * NN * DIMD];
    float dx = row[i * DIMD + 0] - row[j * DIMD + 0];
    float dy = row[i * DIMD + 1] - row[j * DIMD + 1];
    float sq = dx * dx + dy * dy;
    float dist = sqrtf(fmaxf(sq, EPSD));
    acc += expf(bb - dist);
  }
  out[tid] = acc / (float)(S - 1);
}

// ---------------------------------------------------------------------------
// Host-side launch sequencing (graph-capture safe: kernels only)
// ---------------------------------------------------------------------------
static inline void launch_gemm(bool ta, bool tb,
                               const float* A, int lda, int sA,
                               const float* B, int ldb, int sB,
                               float* D, int ldd, int sD,
                               int M, int N, int K, int Z, hipStream_t s) {
  dim3 grid(N / 16, M / 16, Z), blk(32);
  if (ta && !tb)
    wmma_gemm_f32<true, false><<<grid, blk, 0, s>>>(A, lda, sA, B, ldb, sB, D, ldd, sD, K);
  else if (!ta && tb)
    wmma_gemm_f32<false, true><<<grid, blk, 0, s>>>(A, lda, sA, B, ldb, sB, D, ldd, sD, K);
  else if (!ta && !tb)
    wmma_gemm_f32<false, false><<<grid, blk, 0, s>>>(A, lda, sA, B, ldb, sB, D, ldd, sD, K);
  else
    wmma_gemm_f32<true, true><<<grid, blk, 0, s>>>(A, lda, sA, B, ldb, sB, D, ldd, sD, K);
}

extern "C" void kernel_launch(void* const* d_in, const int* in_sizes, int n_in,
                              void* d_out, int out_size, void* d_ws, size_t ws_size,
                              hipStream_t stream) {
  const float* x0    = (const float*)d_in[0];
  const float* v     = (const float*)d_in[1];
  const float* beta  = (const float*)d_in[2];
  const float* sigma = (const float*)d_in[3];
  const float* x0c   = (const float*)d_in[4];
  const float* ls    = (const float*)d_in[5];
  const float* CQ    = (const float*)d_in[6];
  const int*   ssz   = (const int*)d_in[7];
  float* ws  = (float*)d_ws;
  float* out = (float*)d_out;

  // 0) zero workspace
  zero_ws_kernel<<<(WS_FLOATS + 255) / 256, 256, 0, stream>>>(ws);
  // 1) build structured operands
  build_kernel<<<1, 256, 0, stream>>>(x0, v, x0c, ls, CQ, ssz, ws);
  // 2) eigendecompose Bk' (102) and CC^T (32) in parallel workgroups
  jacobi_kernel<<<2, 256, 0, stream>>>(ws);

  // 3) spectral solve + prediction via batched (z = channel) WMMA GEMM chain
  const int SC = PB * NN;  // per-channel stride of PB x 32 mats
  // M1 = U_B^T X           (112x32x112)
  launch_gemm(true,  false, ws + OFF_UB, PB, 0,  ws + OFF_X,  NN, SC,
              ws + OFF_M1, NN, SC, PB, NN, PB, DIMD, stream);
  // Z  = M1 U_C            (112x32x32)
  launch_gemm(false, false, ws + OFF_M1, NN, SC, ws + OFF_UC, NN, 0,
              ws + OFF_Z,  NN, SC, PB, NN, NN, DIMD, stream);
  // W2 = Z * dC / (s2 + dB dC)
  w2_kernel<<<(2 * PB * NN + 255) / 256, 256, 0, stream>>>(sigma, ws);
  // Y2 = U_B W2            (112x32x112)
  launch_gemm(false, false, ws + OFF_UB, PB, 0,  ws + OFF_W2, NN, SC,
              ws + OFF_Y2, NN, SC, PB, NN, PB, DIMD, stream);
  // M2 = Y2 U_C^T          (112x32x32)
  launch_gemm(false, true,  ws + OFF_Y2, NN, SC, ws + OFF_UC, NN, 0,
              ws + OFF_M2, NN, SC, PB, NN, NN, DIMD, stream);
  // P  = Bc^T M2           (128x32x112)
  launch_gemm(true,  false, ws + OFF_BC, PT, 0,  ws + OFF_M2, NN, SC,
              ws + OFF_P,  NN, PT * NN, PT, NN, PB, DIMD, stream);

  // 4) trajectory + 5) intensity integral
  xt_kernel<<<(PT * NN * DIMD + 255) / 256, 256, 0, stream>>>(ssz, ws);
  intensity_kernel<<<1, 1024, 0, stream>>>(beta, ssz, ws, out);
}